// DGCNNFeatureSpace_52965536694408
// MI455X (gfx1250) — compile-verified
//
#include <hip/hip_runtime.h>
#include <float.h>

#define BB 4
#define NN 4096
#define KNN 20
#define EPSV 1e-5f
#define SLOPE 0.2f

typedef float v2f __attribute__((ext_vector_type(2)));
typedef float v8f __attribute__((ext_vector_type(8)));
typedef unsigned int v4u __attribute__((ext_vector_type(4)));
typedef int v4i __attribute__((ext_vector_type(4)));
typedef int v8i __attribute__((ext_vector_type(8)));

__device__ __forceinline__ v8f wmma4(v2f a, v2f b, v8f c) {
  // D = A(16x4) * B(4x16) + C(16x16), all f32
  return __builtin_amdgcn_wmma_f32_16x16x4_f32(
      false, a, false, b, (short)0, c, false, false);
}

// Load a v2f fragment from a row of length LEN at element offset ca.
// LEN % 4 == 0 -> straight b64 load; otherwise clamp+select (no exec branches).
template<int LEN>
__device__ __forceinline__ v2f frag_load(const float* __restrict__ row, int ca) {
  v2f r;
  if constexpr (LEN % 4 == 0) {
    r = *(const v2f*)(row + ca);
  } else {
    int c0 = (ca     < LEN) ? ca     : 0;
    int c1 = (ca + 1 < LEN) ? ca + 1 : 0;
    float v0 = row[c0], v1 = row[c1];
    r.x = (ca     < LEN) ? v0 : 0.f;
    r.y = (ca + 1 < LEN) ? v1 : 0.f;
  }
  return r;
}

// ---------------------------------------------------------------- norms ----
template<int C>
__global__ void norms_kernel(const float* __restrict__ P, float* __restrict__ xx) {
  int i = blockIdx.x * blockDim.x + threadIdx.x;
  if (i >= BB * NN) return;
  const float* p = P + (size_t)i * C;
  float s = 0.f;
#pragma unroll
  for (int c = 0; c < C; ++c) { float v = p[c]; s += v * v; }
  xx[i] = s;
}

// ----------------------------------------------------------------- kNN -----
// One wave per 16-row tile. Gram tile via dual-chained WMMA f32 16x16x4; the
// distance conversion is fused into the tile store (one xx[j] load per tile).
// Per-(row,half) sorted top-20 list lives in LDS; worst cached in a VGPR.
template<int C>
__global__ void __launch_bounds__(32)
knn_kernel(const float* __restrict__ P, const float* __restrict__ xx,
           int* __restrict__ idxo) {
  constexpr int CP4 = (C + 3) / 4;
  __shared__ float tile[256];            // [col][row] column-major 16x16 (distances)
  __shared__ float md[2][16][KNN];
  __shared__ int   mj[2][16][KNN];

  const int b    = blockIdx.y;
  const int i0   = blockIdx.x * 16;
  const int lane = threadIdx.x;
  const int lm   = lane & 15;
  const int half = lane >> 4;
  const int bN   = b * NN;
  const float* Pb = P + (size_t)bN * C;

  for (int t = 0; t < KNN; ++t) { md[half][lm][t] = FLT_MAX; mj[half][lm][t] = 0; }
  float worst = FLT_MAX;

  // A fragments for rows i0..i0+15 (32-bit 16x4 layout: M=lane&15, K=2*half{,+1})
  v2f afr[CP4];
  const float* rowi = Pb + (size_t)(i0 + lm) * C;
#pragma unroll
  for (int cc = 0; cc < CP4; ++cc) afr[cc] = frag_load<C>(rowi, cc * 4 + 2 * half);

  // Row norms for the 8 rows this lane's accumulator VGPRs cover (M=g+8*half)
  float xxi8[8];
#pragma unroll
  for (int g = 0; g < 8; ++g) xxi8[g] = xx[bN + i0 + g + 8 * half];

  for (int j0 = 0; j0 < NN; j0 += 16) {
    const float* rowj = Pb + (size_t)(j0 + lm) * C;
    v8f acc0 = {}, acc1 = {};
#pragma unroll
    for (int cc = 0; cc < CP4; ++cc) {
      v2f bf = frag_load<C>(rowj, cc * 4 + 2 * half);
      if (cc & 1) acc1 = wmma4(afr[cc], bf, acc1);
      else        acc0 = wmma4(afr[cc], bf, acc0);
    }
    v8f acc;
    if constexpr (CP4 > 1) acc = acc0 + acc1; else acc = acc0;

    // This lane stores column N=lm: fold in xx[i] + xx[j] - 2*inner now.
    const float xxj = xx[bN + j0 + lm];
#pragma unroll
    for (int g = 0; g < 8; ++g)
      tile[lm * 16 + half * 8 + g] = xxi8[g] + xxj - 2.f * acc[g];
    __syncthreads();

    // Selection: lane handles row lm, cols [half*8, half*8+8)
#pragma unroll
    for (int t = 0; t < 8; ++t) {
      int c = half * 8 + t;
      int j = j0 + c;
      float d = tile[c * 16 + lm];
      if (d < worst) {
        int q = KNN - 1;
#pragma clang loop unroll(disable)
        while (q > 0 && md[half][lm][q - 1] > d) {
          md[half][lm][q] = md[half][lm][q - 1];
          mj[half][lm][q] = mj[half][lm][q - 1];
          --q;
        }
        md[half][lm][q] = d;
        mj[half][lm][q] = j;
        worst = md[half][lm][KNN - 1];
      }
    }
    __syncthreads();
  }

  // Merge the two per-row sorted half-lists (order of idx is irrelevant
  // downstream, but merge keeps the exact top-20 of the union).
  if (lane < 16) {
    int p0 = 0, p1 = 0;
    int* orow = idxo + (size_t)(bN + i0 + lm) * KNN;
    for (int t = 0; t < KNN; ++t) {
      float d0 = md[0][lm][p0], d1 = md[1][lm][p1];
      int   ja = mj[0][lm][p0], jb = mj[1][lm][p1];
      bool take0 = (d0 < d1) || (d0 == d1 && ja < jb);
      orow[t] = take0 ? ja : jb;
      if (take0) ++p0; else ++p1;
    }
  }
}

// -------------------------------------------------------------- zero -------
__global__ void zero_stats(float* __restrict__ ssum, float* __restrict__ ssq, int O) {
  int i = threadIdx.x;
  if (i < O) { ssum[i] = 0.f; ssq[i] = 0.f; }
}

// ------------------------------------------------------------- pass 1 ------
// Per 16-point tile: centers staged once; per k the 16 neighbor rows are
// gathered into LDS by the Tensor Data Mover (gather mode, 16-bit indices)
// when C is TDM-friendly, then diffed in place. 4 waves split O/16 output
// tiles; dual-chained WMMA over 2C; per-channel sum/sumsq atomics + running
// max/min over k.
template<int C, int O>
__global__ void __launch_bounds__(128)
edge_pass1(const float* __restrict__ P, const int* __restrict__ idxi,
           const float* __restrict__ W,
           float* __restrict__ ssum, float* __restrict__ ssq,
           float* __restrict__ maxy, float* __restrict__ miny) {
  constexpr int C2    = 2 * C;
  constexpr int C2P4  = (C2 + 3) / 4;
  constexpr int NOT_  = O / 16;
  constexpr int NPW   = NOT_ / 4;
  constexpr bool TDM_OK = (C % 4 == 0);

  __shared__ float scen[16 * C];
  __shared__ float snbr[16 * C];
  __shared__ int   sidx[16 * KNN];
  __shared__ int   sidxp[KNN * 8];   // packed 16-bit index pairs for TDM

  const int b    = blockIdx.y;
  const int n0   = blockIdx.x * 16;
  const int tid  = threadIdx.x;
  const int wave = tid >> 5;
  const int lane = tid & 31;
  const int lm   = lane & 15;
  const int half = lane >> 4;
  const int bN   = b * NN;
  const float* Pb = P + (size_t)bN * C;

  if (tid < 64) __builtin_prefetch(W + (size_t)tid * 16, 0, 1);

  for (int e = tid; e < 16 * KNN; e += 128)
    sidx[e] = idxi[(size_t)(bN + n0 + e / KNN) * KNN + (e % KNN)];
  for (int e = tid; e < 16 * C; e += 128)
    scen[e] = Pb[(size_t)(n0 + e / C) * C + (e % C)];
  __syncthreads();
  if constexpr (TDM_OK) {
    for (int e = tid; e < KNN * 8; e += 128) {
      int k = e >> 3, p = e & 7;
      int lo = sidx[(2 * p) * KNN + k];
      int hi = sidx[(2 * p + 1) * KNN + k];
      sidxp[e] = (lo & 0xFFFF) | (hi << 16);
    }
  }

  float vmax[NPW][8], vmin[NPW][8], psum[NPW], psq[NPW];
#pragma unroll
  for (int i = 0; i < NPW; ++i) {
    psum[i] = 0.f; psq[i] = 0.f;
#pragma unroll
    for (int g = 0; g < 8; ++g) { vmax[i][g] = -FLT_MAX; vmin[i][g] = FLT_MAX; }
  }

  for (int k = 0; k < KNN; ++k) {
    __syncthreads();
    bool gathered = false;
    if constexpr (TDM_OK) {
#if defined(__gfx1250__)
      if (tid < 32) {
        // Tensor DMA descriptor: gather-mode 2D tile, 16 x 16-bit row indices
        unsigned lds_off = (unsigned)(size_t)(void*)&snbr[0];
        unsigned long long ga = (unsigned long long)(size_t)(const void*)Pb;
        v4u g0;
        g0[0] = 0x80000001u;                                  // count=1, gather_mode=1, idx16
        g0[1] = lds_off;                                      // lds_addr
        g0[2] = (unsigned)(ga & 0xFFFFFFFFu);                 // global_addr[31:0]
        g0[3] = (unsigned)((ga >> 32) & 0x1FFFFFFu) | 0x80000000u; // [56:32] | type=2
        v8i g1;
        g1[0] = (int)(2u << 16);                              // data_size = 4B
        g1[1] = (int)(((unsigned)C & 0xFFFFu) << 16);         // tensor_dim0 = C
        g1[2] = (int)(((unsigned)NN & 0xFFFFu) << 16);        // tensor_dim1 = N
        g1[3] = (int)((unsigned)C << 16);                     // tile_dim0 = C
        g1[4] = 16;                                           // tile_dim1 = 16 indices
        g1[5] = C;                                            // tensor_dim0_stride = C
        g1[6] = 0;
        g1[7] = 0;
        v4i g2, g3;
        g2[0] = __builtin_amdgcn_readfirstlane(sidxp[k * 8 + 0]);
        g2[1] = __builtin_amdgcn_readfirstlane(sidxp[k * 8 + 1]);
        g2[2] = __builtin_amdgcn_readfirstlane(sidxp[k * 8 + 2]);
        g2[3] = __builtin_amdgcn_readfirstlane(sidxp[k * 8 + 3]);
        g3[0] = __builtin_amdgcn_readfirstlane(sidxp[k * 8 + 4]);
        g3[1] = __builtin_amdgcn_readfirstlane(sidxp[k * 8 + 5]);
        g3[2] = __builtin_amdgcn_readfirstlane(sidxp[k * 8 + 6]);
        g3[3] = __builtin_amdgcn_readfirstlane(sidxp[k * 8 + 7]);
        v8i gz = {};   // unused extra descriptor group (6-arg toolchain form)
        __builtin_amdgcn_tensor_load_to_lds(g0, g1, g2, g3, gz, 0);
        __builtin_amdgcn_s_wait_tensorcnt(0);
      }
      __syncthreads();
      for (int e = tid; e < 16 * C; e += 128) snbr[e] -= scen[e];
      gathered = true;
#endif
    }
    if (!gathered) {
      for (int e = tid; e < 16 * C; e += 128) {
        int r = e / C, c = e % C;
        snbr[e] = Pb[(size_t)sidx[r * KNN + k] * C + c] - scen[e];
      }
    }
    __syncthreads();

#pragma unroll
    for (int oti = 0; oti < NPW; ++oti) {
      const int o0 = (wave * NPW + oti) * 16;
      const float* Wrow = W + (size_t)(o0 + lm) * C2;
      v8f acc0 = {}, acc1 = {};
#pragma unroll
      for (int cc = 0; cc < C2P4; ++cc) {
        const int cbase = cc * 4;
        const int ca = cbase + 2 * half;
        v2f a;
        if constexpr (C % 4 == 0) {
          // chunk never straddles the diff/center boundary (folds after unroll)
          if (cbase < C) a = *(const v2f*)&snbr[lm * C + ca];
          else           a = *(const v2f*)&scen[lm * C + (ca - C >= 0 ? ca - C : 0)];
        } else {
          // per-element select (layer 1, C=3)
          int ch0 = ca, ch1 = ca + 1;
          float x0 = (ch0 < C)  ? snbr[lm * C + (ch0 < C ? ch0 : 0)]
                   : (ch0 < C2) ? scen[lm * C + (ch0 - C >= 0 && ch0 - C < C ? ch0 - C : 0)] : 0.f;
          float x1 = (ch1 < C)  ? snbr[lm * C + (ch1 < C ? ch1 : 0)]
                   : (ch1 < C2) ? scen[lm * C + (ch1 - C >= 0 && ch1 - C < C ? ch1 - C : 0)] : 0.f;
          a.x = x0; a.y = x1;
        }
        v2f bf = frag_load<C2>(Wrow, ca);
        if (cc & 1) acc1 = wmma4(a, bf, acc1);
        else        acc0 = wmma4(a, bf, acc0);
      }
      v8f acc;
      if constexpr (C2P4 > 1) acc = acc0 + acc1; else acc = acc0;
#pragma unroll
      for (int g = 0; g < 8; ++g) {
        float y = acc[g];
        psum[oti] += y;
        psq[oti]  += y * y;
        vmax[oti][g] = fmaxf(vmax[oti][g], y);
        vmin[oti][g] = fminf(vmin[oti][g], y);
      }
    }
  }

#pragma unroll
  for (int oti = 0; oti < NPW; ++oti) {
    const int o0 = (wave * NPW + oti) * 16;
    const int o  = o0 + lm;
    atomicAdd(&ssum[o], psum[oti]);
    atomicAdd(&ssq[o],  psq[oti]);
#pragma unroll
    for (int g = 0; g < 8; ++g) {
      int row = g + 8 * half;  // C/D layout: vgpr g -> M = g + 8*half
      size_t gi = (size_t)(bN + n0 + row) * O + o;
      maxy[gi] = vmax[oti][g];
      miny[gi] = vmin[oti][g];
    }
  }
}

// ------------------------------------------------------------- pass 2 ------
// max_k leaky(BN(y)) == leaky(BN(max_k y)) for gamma>=0 (min_k otherwise)
template<int O>
__global__ void edge_pass2(const float* __restrict__ maxy, const float* __restrict__ miny,
                           const float* __restrict__ ssum, const float* __restrict__ ssq,
                           const float* __restrict__ gamma, const float* __restrict__ beta,
                           float* __restrict__ out) {
  size_t i = (size_t)blockIdx.x * blockDim.x + threadIdx.x;
  if (i >= (size_t)BB * NN * O) return;
  int o = (int)(i % O);
  const float cnt = (float)BB * NN * KNN;
  float mean = ssum[o] / cnt;
  float var  = ssq[o] / cnt - mean * mean;
  float g    = gamma[o];
  float scale = g * rsqrtf(var + EPSV);
  float base = (g >= 0.f) ? maxy[i] : miny[i];
  float y = scale * (base - mean) + beta[o];
  out[i] = (y >= 0.f) ? y : SLOPE * y;
}

// ------------------------------------------------------------- driver ------
template<int C, int O>
static void run_layer(const float* P, const float* W, const float* gamma,
                      const float* beta, float* out, float* xx, int* idx,
                      float* ssum, float* ssq, float* maxy, float* miny,
                      hipStream_t stream) {
  norms_kernel<C><<<(BB * NN + 255) / 256, 256, 0, stream>>>(P, xx);
  knn_kernel<C><<<dim3(NN / 16, BB), 32, 0, stream>>>(P, xx, idx);
  zero_stats<<<1, 512, 0, stream>>>(ssum, ssq, O);
  edge_pass1<C, O><<<dim3(NN / 16, BB), 128, 0, stream>>>(P, idx, W, ssum, ssq, maxy, miny);
  int n2 = (int)(((size_t)BB * NN * O + 255) / 256);
  edge_pass2<O><<<n2, 256, 0, stream>>>(maxy, miny, ssum, ssq, gamma, beta, out);
}

extern "C" void kernel_launch(void* const* d_in, const int* in_sizes, int n_in,
                              void* d_out, int out_size, void* d_ws, size_t ws_size,
                              hipStream_t stream) {
  (void)in_sizes; (void)n_in; (void)out_size; (void)ws_size;
  const float* x  = (const float*)d_in[0];
  const float* W1 = (const float*)d_in[1];
  const float* W2 = (const float*)d_in[2];
  const float* W3 = (const float*)d_in[3];
  const float* W4 = (const float*)d_in[4];
  const float* g1 = (const float*)d_in[5];
  const float* b1 = (const float*)d_in[6];
  const float* g2 = (const float*)d_in[7];
  const float* b2 = (const float*)d_in[8];
  const float* g3 = (const float*)d_in[9];
  const float* b3 = (const float*)d_in[10];
  const float* g4 = (const float*)d_in[11];
  const float* b4 = (const float*)d_in[12];
  float* out = (float*)d_out;

  float* ws   = (float*)d_ws;
  float* xx   = ws;                                   // BB*NN
  float* ssum = xx   + (size_t)BB * NN;               // 256
  float* ssq  = ssum + 256;                           // 256
  float* maxy = ssq  + 256;                           // BB*NN*256
  float* miny = maxy + (size_t)BB * NN * 256;         // BB*NN*256
  float* buf0 = miny + (size_t)BB * NN * 256;         // BB*NN*128
  float* buf1 = buf0 + (size_t)BB * NN * 128;         // BB*NN*64
  int*   idx  = (int*)(buf1 + (size_t)BB * NN * 64);  // BB*NN*KNN ints

  run_layer<3,   64 >(x,    W1, g1, b1, buf0, xx, idx, ssum, ssq, maxy, miny, stream);
  run_layer<64,  64 >(buf0, W2, g2, b2, buf1, xx, idx, ssum, ssq, maxy, miny, stream);
  run_layer<64,  128>(buf1, W3, g3, b3, buf0, xx, idx, ssum, ssq, maxy, miny, stream);
  run_layer<128, 256>(buf0, W4, g4, b4, out,  xx, idx, ssum, ssq, maxy, miny, stream);
}